// RegularizedInference_67559835566282
// MI455X (gfx1250) — compile-verified
//
#include <hip/hip_runtime.h>

#define NDIM 4096
#define SCAS 32
#define NGROUPS 256
#define GAMMA_C 0.1f
#define EPS_C 1e-8f
#define DIAG_STAB_C 1e-3f

typedef __attribute__((ext_vector_type(2))) float v2f;
typedef __attribute__((ext_vector_type(8))) float v8f;

// ---------------------------------------------------------------------------
// Pack cascade column j (over the 32 cascades) into a 32-bit mask.
__global__ void reg_packbits(const int* __restrict__ casc, unsigned* __restrict__ bits) {
  int j = blockIdx.x * blockDim.x + threadIdx.x;
  unsigned b = 0u;
  #pragma unroll
  for (int s = 0; s < SCAS; ++s)
    b |= ((unsigned)(casc[s * NDIM + j] & 1)) << s;
  bits[j] = b;
}

// Group member counts.
__global__ void reg_counts(const int* __restrict__ ids, int* __restrict__ counts) {
  int i = blockIdx.x * blockDim.x + threadIdx.x;
  atomicAdd(&counts[ids[i]], 1);
}

// ---------------------------------------------------------------------------
// One block per row i: compute A_prob, write -A_prob into W (off-diag Laplacian),
// accumulate term1 (popcount-weighted log), sum of squares, and per-group sums.
__global__ void reg_phase1(const float* __restrict__ A, const float* __restrict__ P,
                           const unsigned* __restrict__ bits, const int* __restrict__ ids,
                           float* __restrict__ W, float* __restrict__ gsums,
                           float* __restrict__ acc) {
  const int i = blockIdx.x;
  const int t = threadIdx.x;
  const unsigned bi = bits[i];
  const int g = ids[i];
  const size_t rowoff = (size_t)i * NDIM;
  float t1 = 0.f, sq = 0.f;
  for (int j = t; j < NDIM; j += 256) {
    float a = A[rowoff + j];
    float p = P[rowoff + j];
    float ap = (i == j) ? 0.f : (1.f / (1.f + expf(-a))) * p;
    W[rowoff + j] = -ap;
    atomicAdd(&gsums[(size_t)g * NDIM + j], ap);
    float m = logf(1.f - ap + EPS_C);
    t1 += m * (float)__popc(bi & ~bits[j]);
    sq += ap * ap;
  }
  __shared__ float red0[256];
  __shared__ float red1[256];
  red0[t] = t1; red1[t] = sq;
  __syncthreads();
  for (int s = 128; s > 0; s >>= 1) {
    if (t < s) { red0[t] += red0[t + s]; red1[t] += red1[t + s]; }
    __syncthreads();
  }
  if (t == 0) { atomicAdd(&acc[0], red0[0]); atomicAdd(&acc[1], red1[0]); }
}

// Column sums -> Laplacian diagonal (+ stabilizer). Coalesced: thread = column.
__global__ void reg_coldiag(float* __restrict__ W) {
  int j = blockIdx.x * blockDim.x + threadIdx.x;
  float s = 0.f;
  for (int i = 0; i < NDIM; ++i) s += W[(size_t)i * NDIM + j];
  W[(size_t)j * NDIM + j] = -s + DIAG_STAB_C;   // stored off-diags are -A_prob
}

// Omega second term: sum_g,j S^2 / max(cnt,1).
__global__ void reg_omega2(const float* __restrict__ gsums, const int* __restrict__ counts,
                           float* __restrict__ acc) {
  int idx = blockIdx.x * blockDim.x + threadIdx.x;
  int g = idx >> 12;                 // / NDIM
  float s = gsums[idx];
  int c = counts[g];
  float v = s * s / (float)(c < 1 ? 1 : c);
  __shared__ float red[256];
  red[threadIdx.x] = v;
  __syncthreads();
  for (int st = 128; st > 0; st >>= 1) {
    if (threadIdx.x < st) red[threadIdx.x] += red[threadIdx.x + st];
    __syncthreads();
  }
  if (threadIdx.x == 0) atomicAdd(&acc[2], red[0]);
}

// ---------------------------------------------------------------------------
// Unpivoted LU of the 16x16 diagonal block (W is column diagonally dominant =>
// stable without pivoting), logdet accumulation, and explicit inverses of the
// unit-lower L11 and upper U11 factors for the WMMA panel solves.
__global__ void reg_panel_lu(float* __restrict__ W, float* __restrict__ invL,
                             float* __restrict__ invU, float* __restrict__ acc, int k) {
  __shared__ float D[16][17];
  __shared__ float Li[16][17];
  __shared__ float Ui[16][17];
  const int t = threadIdx.x;
  const int r = t >> 4, c = t & 15;
  D[r][c] = W[(size_t)(k + r) * NDIM + (k + c)];
  __syncthreads();
  for (int p = 0; p < 16; ++p) {
    if (r > p && c == p) D[r][p] = D[r][p] / D[p][p];
    __syncthreads();
    if (r > p && c > p) D[r][c] -= D[r][p] * D[p][c];
    __syncthreads();
  }
  W[(size_t)(k + r) * NDIM + (k + c)] = D[r][c];
  if (t == 0) {
    float ld = 0.f;
    for (int p = 0; p < 16; ++p) ld += logf(fabsf(D[p][p]));
    atomicAdd(&acc[3], ld);
  }
  if (t < 16) {                      // invert unit-lower L, column t
    const int j = t;
    float x[16];
    for (int i2 = 0; i2 < 16; ++i2) {
      float v = (i2 == j) ? 1.f : 0.f;
      for (int p = j; p < i2; ++p) v -= D[i2][p] * x[p];
      x[i2] = v;
    }
    for (int i2 = 0; i2 < 16; ++i2) Li[i2][j] = (i2 < j) ? 0.f : x[i2];
  } else if (t < 32) {               // invert upper U, column t-16
    const int j = t - 16;
    float x[16];
    for (int i2 = 15; i2 >= 0; --i2) {
      float v = (i2 == j) ? 1.f : 0.f;
      for (int p = i2 + 1; p <= j; ++p) v -= D[i2][p] * x[p];
      x[i2] = (i2 <= j) ? v / D[i2][i2] : 0.f;
    }
    for (int i2 = 0; i2 < 16; ++i2) Ui[i2][j] = x[i2];
  }
  __syncthreads();
  invL[r * 16 + c] = Li[r][c];
  invU[r * 16 + c] = Ui[r][c];
}

// L21 = A21 * U11^-1 : one 16x16 tile per wave, K=16 via 4x V_WMMA_F32_16X16X4_F32.
__global__ void reg_trsm_l21(float* __restrict__ W, const float* __restrict__ invU,
                             int k, int T) {
  const int wave = threadIdx.x >> 5, lane = threadIdx.x & 31;
  const int tile = blockIdx.x * 8 + wave;
  if (tile * 16 >= T) return;                 // wave-uniform exit; EXEC full for WMMA
  const int row0 = k + 16 + tile * 16;
  const int half = lane >> 4, q = lane & 15;
  v8f accv = {};
  #pragma unroll
  for (int kk = 0; kk < 16; kk += 4) {
    v2f a, b;
    a.x = W[(size_t)(row0 + q) * NDIM + (k + kk + 2 * half)];
    a.y = W[(size_t)(row0 + q) * NDIM + (k + kk + 2 * half + 1)];
    b.x = invU[(kk + 2 * half) * 16 + q];
    b.y = invU[(kk + 2 * half + 1) * 16 + q];
    accv = __builtin_amdgcn_wmma_f32_16x16x4_f32(false, a, false, b, (short)0, accv,
                                                 false, false);
  }
  #pragma unroll
  for (int v = 0; v < 8; ++v)
    W[(size_t)(row0 + v + 8 * half) * NDIM + (k + q)] = accv[v];
}

// U12 = L11^-1 * A12 : one 16x16 tile per wave.
__global__ void reg_trsm_u12(float* __restrict__ W, const float* __restrict__ invL,
                             int k, int T) {
  const int wave = threadIdx.x >> 5, lane = threadIdx.x & 31;
  const int tile = blockIdx.x * 8 + wave;
  if (tile * 16 >= T) return;
  const int col0 = k + 16 + tile * 16;
  const int half = lane >> 4, q = lane & 15;
  v8f accv = {};
  #pragma unroll
  for (int kk = 0; kk < 16; kk += 4) {
    v2f a, b;
    a.x = invL[q * 16 + kk + 2 * half];
    a.y = invL[q * 16 + kk + 2 * half + 1];
    b.x = W[(size_t)(k + kk + 2 * half) * NDIM + col0 + q];
    b.y = W[(size_t)(k + kk + 2 * half + 1) * NDIM + col0 + q];
    accv = __builtin_amdgcn_wmma_f32_16x16x4_f32(false, a, false, b, (short)0, accv,
                                                 false, false);
  }
  #pragma unroll
  for (int v = 0; v < 8; ++v)
    W[(size_t)(k + v + 8 * half) * NDIM + col0 + q] = accv[v];
}

// Trailing update A22 -= L21*U12. 64x64 block tile through LDS, 8 waves,
// each wave owns two 16x16 accumulators; product negated via -A fragment so
// WMMA's D = A*B + C performs the subtraction in one pass.
__global__ void reg_lu_update(float* __restrict__ W, int k) {
  const int t = threadIdx.x;
  const int lane = t & 31, wave = t >> 5;
  const int half = lane >> 4, q = lane & 15;
  const int r0 = k + 16 + blockIdx.x * 64;
  const int c0 = k + 16 + blockIdx.y * 64;
  __shared__ float Ash[64][17];
  __shared__ float Bsh[16][64];
  for (int idx = t; idx < 64 * 16; idx += 256) {
    int r = idx >> 4, c = idx & 15;
    Ash[r][c] = (r0 + r < NDIM) ? W[(size_t)(r0 + r) * NDIM + (k + c)] : 0.f;
  }
  for (int idx = t; idx < 16 * 64; idx += 256) {
    int r = idx >> 6, c = idx & 63;
    Bsh[r][c] = (c0 + c < NDIM) ? W[(size_t)(k + r) * NDIM + (c0 + c)] : 0.f;
  }
  __syncthreads();
  const int tr = wave >> 1;
  const int tc = (wave & 1) * 2;
  const int row = r0 + tr * 16;
  const int colA = c0 + tc * 16;
  const int colB = c0 + (tc + 1) * 16;
  v8f acc0 = {}, acc1 = {};
  #pragma unroll
  for (int v = 0; v < 8; ++v) {
    int rr = row + v + 8 * half;
    if (rr < NDIM) {
      if (colA + q < NDIM) acc0[v] = W[(size_t)rr * NDIM + colA + q];
      if (colB + q < NDIM) acc1[v] = W[(size_t)rr * NDIM + colB + q];
    }
  }
  #pragma unroll
  for (int kk = 0; kk < 16; kk += 4) {
    v2f a, b0, b1;
    a.x = -Ash[tr * 16 + q][kk + 2 * half];
    a.y = -Ash[tr * 16 + q][kk + 2 * half + 1];
    b0.x = Bsh[kk + 2 * half][tc * 16 + q];
    b0.y = Bsh[kk + 2 * half + 1][tc * 16 + q];
    b1.x = Bsh[kk + 2 * half][(tc + 1) * 16 + q];
    b1.y = Bsh[kk + 2 * half + 1][(tc + 1) * 16 + q];
    acc0 = __builtin_amdgcn_wmma_f32_16x16x4_f32(false, a, false, b0, (short)0, acc0,
                                                 false, false);
    acc1 = __builtin_amdgcn_wmma_f32_16x16x4_f32(false, a, false, b1, (short)0, acc1,
                                                 false, false);
  }
  #pragma unroll
  for (int v = 0; v < 8; ++v) {
    int rr = row + v + 8 * half;
    if (rr < NDIM) {
      if (colA + q < NDIM) W[(size_t)rr * NDIM + colA + q] = acc0[v];
      if (colB + q < NDIM) W[(size_t)rr * NDIM + colB + q] = acc1[v];
    }
  }
}

__global__ void reg_finalize(const float* __restrict__ acc, float* __restrict__ out) {
  float term1 = acc[0], sumsq = acc[1], part2 = acc[2], ld = acc[3];
  float logL = (term1 + (float)SCAS * ld) / ((float)SCAS * (float)NDIM);
  out[0] = -logL + GAMMA_C * (sumsq - part2);
}

// ---------------------------------------------------------------------------
extern "C" void kernel_launch(void* const* d_in, const int* in_sizes, int n_in,
                              void* d_out, int out_size, void* d_ws, size_t ws_size,
                              hipStream_t stream) {
  const float* A    = (const float*)d_in[0];
  const float* P    = (const float*)d_in[1];
  const int*   casc = (const int*)d_in[2];
  const int*   ids  = (const int*)d_in[3];
  float* out = (float*)d_out;

  char* ws = (char*)d_ws;
  const size_t offW      = 0;
  const size_t offSums   = offW + (size_t)NDIM * NDIM * sizeof(float);      // 64 MB
  const size_t offCounts = offSums + (size_t)NGROUPS * NDIM * sizeof(float);// +4 MB
  const size_t offAcc    = offCounts + NGROUPS * sizeof(int);               // +1 KB
  const size_t offBits   = offAcc + 64;                                     // 4 scalars
  const size_t offInvL   = offBits + NDIM * sizeof(unsigned);               // +16 KB
  const size_t offInvU   = offInvL + 256 * sizeof(float);

  float*    W      = (float*)(ws + offW);
  float*    gsums  = (float*)(ws + offSums);
  int*      counts = (int*)(ws + offCounts);
  float*    acc    = (float*)(ws + offAcc);
  unsigned* bits   = (unsigned*)(ws + offBits);
  float*    invL   = (float*)(ws + offInvL);
  float*    invU   = (float*)(ws + offInvU);

  // Zero group sums, counts, and the 4 scalar accumulators (capture-safe).
  hipMemsetAsync(ws + offSums,
                 0, (size_t)NGROUPS * NDIM * sizeof(float) + NGROUPS * sizeof(int) + 64,
                 stream);

  reg_packbits<<<NDIM / 256, 256, 0, stream>>>(casc, bits);
  reg_counts<<<NDIM / 256, 256, 0, stream>>>(ids, counts);
  reg_phase1<<<NDIM, 256, 0, stream>>>(A, P, bits, ids, W, gsums, acc);
  reg_coldiag<<<NDIM / 256, 256, 0, stream>>>(W);
  reg_omega2<<<(NGROUPS * NDIM) / 256, 256, 0, stream>>>(gsums, counts, acc);

  for (int kb = 0; kb < NDIM / 16; ++kb) {
    const int k = kb * 16;
    reg_panel_lu<<<1, 256, 0, stream>>>(W, invL, invU, acc, k);
    const int T = NDIM - k - 16;
    if (T > 0) {
      const int rblocks = (T + 127) / 128;   // 8 tiles of 16 rows/cols per block
      reg_trsm_l21<<<rblocks, 256, 0, stream>>>(W, invU, k, T);
      reg_trsm_u12<<<rblocks, 256, 0, stream>>>(W, invL, k, T);
      const int g64 = (T + 63) / 64;
      dim3 grid(g64, g64);
      reg_lu_update<<<grid, 256, 0, stream>>>(W, k);
    }
  }
  reg_finalize<<<1, 1, 0, stream>>>(acc, out);
}